// ResidualVectorQuantize_44796508897705
// MI455X (gfx1250) — compile-verified
//
#include <hip/hip_runtime.h>

// RVQ fused kernel for MI455X (gfx1250, wave32, WMMA).
// One wave = one (batch, 16-column) tile; residual lives in LDS across all 9
// quantizer stages; all GEMMs via v_wmma_f32_16x16x4_f32 (f32 precision).
// Large streaming outputs use non-temporal store hints to keep the hot
// weight/codebook working set resident in L2.

typedef __attribute__((ext_vector_type(2))) float v2f;
typedef __attribute__((ext_vector_type(4))) float v4f;
typedef __attribute__((ext_vector_type(8))) float v8f;

namespace {
constexpr int kB    = 16;
constexpr int kD    = 512;
constexpr int kT    = 2048;
constexpr int kNQ   = 9;
constexpr int kCS   = 1024;
constexpr int kCD   = 8;
constexpr int kTile = 16;
constexpr int kRS   = 20;   // padded LDS row stride (floats): 16B-aligned rows, conflict-free
}

__device__ __forceinline__ v8f wmma4(v2f a, v2f b, v8f c) {
  // D = A(16x4 f32) * B(4x16 f32) + C(16x16 f32)
  return __builtin_amdgcn_wmma_f32_16x16x4_f32(false, a, false, b, (short)0, c, false, false);
}

__device__ __forceinline__ void nt_store(float* p, float v) {
  __builtin_nontemporal_store(v, p);
}
__device__ __forceinline__ void nt_store4(float* p, v4f v) {
  __builtin_nontemporal_store(v, (v4f*)p);
}

// --- setup: weight-norm in_proj rows (norm over D=512), one wave per row ---
__global__ __launch_bounds__(32) void rvq_wn_in_kernel(const float* __restrict__ in_v,
                                                       const float* __restrict__ in_g,
                                                       float* __restrict__ win) {
  const int row  = blockIdx.x;            // s*8 + c  (72 rows)
  const int lane = threadIdx.x;
  const float* v = in_v + (size_t)row * kD;
  float ss = 0.f;
  for (int i = lane; i < kD; i += 32) { const float x = v[i]; ss += x * x; }
#pragma unroll
  for (int off = 16; off > 0; off >>= 1) ss += __shfl_xor(ss, off, 32);
  const float sc = in_g[row] / sqrtf(ss);
  for (int i = lane; i < kD; i += 32) win[(size_t)row * kD + i] = v[i] * sc;
}

// --- setup: weight-norm out_proj rows (norm over 8) + L2-normalized codebook ---
__global__ void rvq_wn_out_cn_kernel(const float* __restrict__ out_v,
                                     const float* __restrict__ out_g,
                                     const float* __restrict__ cb,
                                     float* __restrict__ wout,
                                     float* __restrict__ cn) {
  const int i     = blockIdx.x * blockDim.x + threadIdx.x;
  const int nWout = kNQ * kD;    // 4608
  const int nCn   = kNQ * kCS;   // 9216
  if (i < nWout) {
    const float* v = out_v + (size_t)i * kCD;
    float ss = 0.f;
#pragma unroll
    for (int c = 0; c < kCD; ++c) ss += v[c] * v[c];
    const float sc = out_g[i] / sqrtf(ss);
#pragma unroll
    for (int c = 0; c < kCD; ++c) wout[(size_t)i * kCD + c] = v[c] * sc;
  } else if (i < nWout + nCn) {
    const int j    = i - nWout;
    const float* v = cb + (size_t)j * kCD;
    float ss = 0.f;
#pragma unroll
    for (int c = 0; c < kCD; ++c) ss += v[c] * v[c];
    const float sc = 1.0f / fmaxf(sqrtf(ss), 1e-12f);
#pragma unroll
    for (int c = 0; c < kCD; ++c) cn[(size_t)j * kCD + c] = v[c] * sc;
  }
}

__global__ __launch_bounds__(32) void rvq_main_kernel(
    const float* __restrict__ z,
    const float* __restrict__ in_b,
    const float* __restrict__ out_b,
    const float* __restrict__ cb_raw,
    const float* __restrict__ win,
    const float* __restrict__ wout,
    const float* __restrict__ cn,
    float* __restrict__ out_codes,
    float* __restrict__ out_zO,
    float* __restrict__ out_zi,
    float* __restrict__ out_zq,
    float* __restrict__ out_zo) {
  __shared__ float R[kD * kRS];     // residual tile, R[d*kRS + t]
  __shared__ float ZI[kCD * kTile]; // z_i staging (C-layout -> B-layout bridge)
  __shared__ int   IDX[kTile];      // per-column code index

  const int lane = threadIdx.x;
  const int m    = lane & 15;           // A: row M / B,C,D: column N
  const int kh   = (lane >> 4) << 1;    // K base per lane-half (0 or 2)
  const int hi8  = (lane >> 4) << 3;    // C/D row offset per lane-half (0 or 8)

  const int tilesPerB = kT / kTile;     // 128
  const int b  = blockIdx.x / tilesPerB;
  const int t0 = (blockIdx.x % tilesPerB) * kTile;

  const float* zb = z + (size_t)b * kD * kT + t0;

  // ---- stage residual tile into LDS (float4, coalesced; RT so reload hits L2) ----
  const int r4 = lane >> 2;
  const int c4 = (lane & 3) << 2;
  for (int d0 = 0; d0 < kD; d0 += 8) {
    const int d = d0 + r4;
    const v4f v4 = *(const v4f*)(zb + (size_t)d * kT + c4);
    *(v4f*)(&R[d * kRS + c4]) = v4;
  }
  __syncthreads();

  for (int s = 0; s < kNQ; ++s) {
    const float* winS  = win    + (size_t)s * kCD * kD;
    const float* cnS   = cn     + (size_t)s * kCS * kCD;
    const float* cbS   = cb_raw + (size_t)s * kCS * kCD;
    const float* woutS = wout   + (size_t)s * kD * kCD;
    const float* ibS   = in_b  + s * kCD;
    const float* obS   = out_b + s * kD;
    const size_t zsb   = (size_t)b * kNQ + s;

    // ---- in_proj: z_i(8x16) = Win(8x512) @ R(512x16), K chunks of 4 ----
    v8f acc = {0.f, 0.f, 0.f, 0.f, 0.f, 0.f, 0.f, 0.f};
    for (int d0 = 0; d0 < kD; d0 += 4) {
      v2f a, bv;
      // A rows m>=8 read adjacent workspace (in-bounds) and land in unused C rows.
      a.x  = winS[m * kD + d0 + kh + 0];
      a.y  = winS[m * kD + d0 + kh + 1];
      bv.x = R[(d0 + kh + 0) * kRS + m];
      bv.y = R[(d0 + kh + 1) * kRS + m];
      acc  = wmma4(a, bv, acc);
    }

    // ---- bias, emit z_i_s (NT), stage z_i into LDS ----
    if (lane < 16) {
      float* zi_out = out_zi + zsb * kCD * kT + t0 + m;
#pragma unroll
      for (int c = 0; c < kCD; ++c) {
        const float v = acc[c] + ibS[c];
        ZI[c * kTile + m] = v;
        nt_store(zi_out + (size_t)c * kT, v);
      }
    }
    __syncthreads();

    // ---- codebook search: argmax_j  cn_hat(1024x8) @ z_i(8x16) ----
    // (enc normalization is a positive per-column scale -> argmax-invariant)
    v2f e0, e1;
    e0.x = ZI[(0 + kh) * kTile + m];
    e0.y = ZI[(1 + kh) * kTile + m];
    e1.x = ZI[(4 + kh) * kTile + m];
    e1.y = ZI[(5 + kh) * kTile + m];

    float bestv = -3.402823466e38f;
    int   besti = 0;
    for (int j0 = 0; j0 < kCS; j0 += 16) {
      const float* crow = cnS + (size_t)(j0 + m) * kCD;
      v2f a0, a1;
      a0.x = crow[kh + 0];      a0.y = crow[kh + 1];
      a1.x = crow[4 + kh + 0];  a1.y = crow[4 + kh + 1];
      v8f sc = {0.f, 0.f, 0.f, 0.f, 0.f, 0.f, 0.f, 0.f};
      sc = wmma4(a0, e0, sc);
      sc = wmma4(a1, e1, sc);
#pragma unroll
      for (int r = 0; r < 8; ++r) {
        const float v  = sc[r];
        const int code = j0 + r + hi8;
        if (v > bestv) { bestv = v; besti = code; }
      }
    }
    {   // merge the two lane-halves (same t, different code subsets)
      const float ov = __shfl_xor(bestv, 16, 32);
      const int   oi = __shfl_xor(besti, 16, 32);
      if (ov > bestv || (ov == bestv && oi < besti)) { bestv = ov; besti = oi; }
    }
    if (lane < 16) {
      IDX[m] = besti;
      out_codes[zsb * kT + t0 + m] = (float)besti;
    }
    __syncthreads();

    // ---- gather z_q (straight-through), emit z_q_s (NT), build B operand ----
    v2f q0, q1;
    {
      const float* qrow = cbS + (size_t)IDX[m] * kCD;
      const float z0 = ZI[(0 + kh) * kTile + m];
      const float z1 = ZI[(1 + kh) * kTile + m];
      const float z4 = ZI[(4 + kh) * kTile + m];
      const float z5 = ZI[(5 + kh) * kTile + m];
      q0.x = z0 + (qrow[kh + 0] - z0);       // mimic z_i + (z_q_raw - z_i)
      q0.y = z1 + (qrow[kh + 1] - z1);
      q1.x = z4 + (qrow[4 + kh + 0] - z4);
      q1.y = z5 + (qrow[4 + kh + 1] - z5);
      float* zq_out = out_zq + zsb * kCD * kT + t0 + m;
      nt_store(zq_out + (size_t)(0 + kh) * kT, q0.x);  // halves cover disjoint c rows
      nt_store(zq_out + (size_t)(1 + kh) * kT, q0.y);
      nt_store(zq_out + (size_t)(4 + kh) * kT, q1.x);
      nt_store(zq_out + (size_t)(5 + kh) * kT, q1.y);
    }

    // ---- out_proj: z_o(512x16) = Wout(512x8) @ z_q(8x16); emit (NT) + update R ----
    float* zo_out = out_zo + zsb * (size_t)kD * kT + t0 + m;
    for (int d0 = 0; d0 < kD; d0 += 16) {
      const float* wrow = woutS + (size_t)(d0 + m) * kCD;
      v2f a0, a1;
      a0.x = wrow[kh + 0];      a0.y = wrow[kh + 1];
      a1.x = wrow[4 + kh + 0];  a1.y = wrow[4 + kh + 1];
      v8f zo = {0.f, 0.f, 0.f, 0.f, 0.f, 0.f, 0.f, 0.f};
      zo = wmma4(a0, q0, zo);
      zo = wmma4(a1, q1, zo);
#pragma unroll
      for (int r = 0; r < 8; ++r) {
        const int dr  = d0 + r + hi8;
        const float v = zo[r] + obS[dr];
        nt_store(zo_out + (size_t)dr * kT, v);
        R[dr * kRS + m] -= v;
      }
    }
    __syncthreads();
  }

  // ---- z_O = z - final residual (exact: z_O accumulated all z_o) ----
  float* zOb = out_zO + (size_t)b * kD * kT + t0;
  for (int d0 = 0; d0 < kD; d0 += 8) {
    const int d = d0 + r4;
    const v4f zv = *(const v4f*)(zb + (size_t)d * kT + c4);   // RT: likely L2 hit
    const v4f rv = *(const v4f*)(&R[d * kRS + c4]);
    nt_store4(zOb + (size_t)d * kT + c4, zv - rv);
  }
}

extern "C" void kernel_launch(void* const* d_in, const int* in_sizes, int n_in,
                              void* d_out, int out_size, void* d_ws, size_t ws_size,
                              hipStream_t stream) {
  (void)in_sizes; (void)n_in; (void)out_size; (void)ws_size;
  const float* z     = (const float*)d_in[0];
  const float* in_v  = (const float*)d_in[1];
  const float* in_g  = (const float*)d_in[2];
  const float* in_b  = (const float*)d_in[3];
  const float* out_v = (const float*)d_in[4];
  const float* out_g = (const float*)d_in[5];
  const float* out_b = (const float*)d_in[6];
  const float* cb    = (const float*)d_in[7];

  // workspace: 147456 floats = 576 KB
  float* ws   = (float*)d_ws;
  float* win  = ws;             // 9*8*512   = 36864
  float* wout = ws + 36864;     // 9*512*8   = 36864
  float* cn   = ws + 73728;     // 9*1024*8  = 73728

  rvq_wn_in_kernel<<<dim3(kNQ * kCD), dim3(32), 0, stream>>>(in_v, in_g, win);
  {
    const int rows = kNQ * kD + kNQ * kCS;   // 13824
    rvq_wn_out_cn_kernel<<<dim3((rows + 255) / 256), dim3(256), 0, stream>>>(
        out_v, out_g, cb, wout, cn);
  }

  // output layout (flat concat, return order): codes, z_O, z_i_s, z_q_s, z_o_s
  float* out     = (float*)d_out;
  float* o_codes = out;                          // 16*9*2048     = 294912
  float* o_zO    = out + 294912;                 // 16*512*2048   = 16777216
  float* o_zi    = out + 294912 + 16777216;      // 16*9*8*2048   = 2359296
  float* o_zq    = o_zi + 2359296;               // 2359296
  float* o_zo    = o_zq + 2359296;               // 16*9*512*2048

  rvq_main_kernel<<<dim3(kB * (kT / kTile)), dim3(32), 0, stream>>>(
      z, in_b, out_b, cb, win, wout, cn, o_codes, o_zO, o_zi, o_zq, o_zo);
}